// CooccurrenceMatrix_59777354825861
// MI455X (gfx1250) — compile-verified
//
#include <hip/hip_runtime.h>

// Problem constants (from reference): 64 rows, 786432 px/row, 256 levels.
#define LEVELS 256
#define WAVES_PER_BLOCK 8            // 256 threads / wave32

typedef int   v4i __attribute__((ext_vector_type(4)));
typedef float v2f __attribute__((ext_vector_type(2)));
typedef float v4f __attribute__((ext_vector_type(4)));
typedef float v8f __attribute__((ext_vector_type(8)));

// ---------------------------------------------------------------------------
// Kernel 1: zero the per-row histogram accumulator in workspace (u32 counts).
// Must run every call: harness does not re-poison/zero d_ws between replays.
// ---------------------------------------------------------------------------
__global__ void hist_zero_kernel(unsigned* __restrict__ g) {
    g[blockIdx.x * blockDim.x + threadIdx.x] = 0u;
}

// ---------------------------------------------------------------------------
// Kernel 2: per-row histogram (the bandwidth-dominant phase: 201 MB streamed).
//   grid = (blocks_per_row, B). Each block covers a contiguous chunk of row b.
//   Per-WAVE private LDS histograms (8 x 256 u32 = 8 KB << 320 KB WGP LDS)
//   -> no inter-wave LDS atomic contention (ds_add_u32, no return).
//   Input streamed with 128-bit non-temporal loads (read-once data; keep it
//   from evicting useful lines in the 192 MB L2). u32 counts -> exact and
//   order-independent -> deterministic.
// ---------------------------------------------------------------------------
__global__ __launch_bounds__(256) void hist_kernel(const int* __restrict__ x,
                                                   unsigned* __restrict__ ghist,
                                                   int N, int chunk) {
    __shared__ unsigned lh[WAVES_PER_BLOCK * LEVELS];
    const int t = threadIdx.x;
    const int b = blockIdx.y;

    #pragma unroll
    for (int i = 0; i < WAVES_PER_BLOCK; ++i) lh[i * LEVELS + t] = 0u;
    __syncthreads();

    unsigned* wh = &lh[(t >> 5) * LEVELS];       // this wave's private copy

    const long long rowBase = (long long)b * (long long)N;
    const v4i* __restrict__ p = (const v4i*)(x + rowBase);

    int start = blockIdx.x * chunk;
    int end   = start + chunk;
    if (end > N) end = N;

    // 256 threads x 4 elements = 1024 elements per sweep; chunk is a
    // multiple of 1024 so every b128 load is 16B-aligned and in-bounds.
    for (int i = start + t * 4; i < end; i += 256 * 4) {
        v4i v = __builtin_nontemporal_load(&p[i >> 2]);
        atomicAdd(&wh[v.x & 255], 1u);           // ds_add_u32
        atomicAdd(&wh[v.y & 255], 1u);
        atomicAdd(&wh[v.z & 255], 1u);
        atomicAdd(&wh[v.w & 255], 1u);
    }
    __syncthreads();

    // Reduce the 8 wave-private copies; one global atomic per bin per block.
    unsigned s = 0;
    #pragma unroll
    for (int w = 0; w < WAVES_PER_BLOCK; ++w) s += lh[w * LEVELS + t];
    if (s) atomicAdd(&ghist[b * LEVELS + t], s); // global_atomic_add_u32
}

// ---------------------------------------------------------------------------
// Kernel 3: broadcast h[b, i] across 256 columns via the matrix pipe.
//   out[b, i, j] = h[b, i] is the rank-1 product A x B with A[:,0]=h,
//   B[0,:]=1. One V_WMMA_F32_16X16X4_F32 per wave builds a 16x16 tile whose
//   every row is constant. f32 WMMA keeps counts (<= 786432 < 2^24) exact.
//
//   ISA VGPR layouts (CDNA5 ISA):
//     A 16x4 f32 : lane = M; v0 = {K0 | K2}, v1 = {K1 | K3}
//     B 4x16 f32 : v0 lanes0-15 = (K0, N=lane), lanes16-31 = (K2, ...)
//     D 16x16 f32: VGPR r: lane<16 -> (M=r, N=lane); lane>=16 -> (M=r+8, N=lane-16)
//
//   Store phase: rows are constant, so each lane replicates its d[r] into a
//   float4 and writes b128 (512 B/instr/wave vs 128 B for b32): lane group
//   {0..15 -> row r, 16..31 -> row r+8}, lane covers cols 4*(L&15)+64*j,
//   j=0..3. 32 b128 stores per wave, all immediate offsets off one base
//   (r*1024 + j*256 bytes) -> single store clause.
//   grid = (4, B), block = 128 (4 waves); EXEC all-ones at the WMMA.
// ---------------------------------------------------------------------------
__global__ __launch_bounds__(128) void bcast_wmma_kernel(const unsigned* __restrict__ ghist,
                                                         float* __restrict__ out) {
    const int b     = blockIdx.y;
    const int lane  = threadIdx.x & 31;
    const int wave  = threadIdx.x >> 5;
    const int mbase = (blockIdx.x * 4 + wave) * 16;   // M-tile origin, 0..240

    const unsigned* __restrict__ h = ghist + b * LEVELS;

    // All lanes issue the load (no divergence); lanes >= 16 zero their A value.
    unsigned cnt = h[mbase + (lane & 15)];
    float hv = (lane < 16) ? (float)cnt : 0.0f;

    v2f a;  a.x  = hv;                             // A[m,0] = h[mbase+m]
            a.y  = 0.0f;                           // A[m,1] = A[m,3] = 0 (K2 lanes>=16 also 0)
    v2f bm; bm.x = (lane < 16) ? 1.0f : 0.0f;      // B[0,n] = 1, B[2,n] = 0
            bm.y = 0.0f;                           // B[1,n] = B[3,n] = 0
    v8f c = {};

    // D[m,n] = sum_k A[m,k]*B[k,n] = h[mbase+m]  for all n
    v8f d = __builtin_amdgcn_wmma_f32_16x16x4_f32(
        /*neg_a=*/false, a, /*neg_b=*/false, bm,
        /*c_mod=*/(short)0, c, /*reuse_a=*/false, /*reuse_b=*/false);

    float* __restrict__ outb = out + ((size_t)b << 16);   // b * 256 * 256
    const int n  = lane & 15;
    const int mh = (lane >> 4) << 3;                      // 0 for lanes<16, 8 for lanes>=16

    // Per-lane base: row (mbase+mh), column 4*n. All further addressing is
    // immediate: row step = 1024 B, column-tile step = 256 B.
    float* base = outb + (size_t)(mbase + mh) * 256 + n * 4;

    #pragma unroll
    for (int r = 0; r < 8; ++r) {
        v4f v;                                     // row is constant -> replicate
        v.x = d[r]; v.y = d[r]; v.z = d[r]; v.w = d[r];
        v4f* rp = (v4f*)(base + (size_t)r * 256);  // row mbase+mh+r
        #pragma unroll
        for (int j = 0; j < 4; ++j) {
            rp[j * 16] = v;                        // cols 4n + 64j .. +3
        }
    }
}

// ---------------------------------------------------------------------------
// kernel_launch: inputs -> x (int32, [B, N]); out (f32, [B, 256, 256]);
// d_ws holds the B*256 u32 histogram accumulator.
// ---------------------------------------------------------------------------
extern "C" void kernel_launch(void* const* d_in, const int* in_sizes, int n_in,
                              void* d_out, int out_size, void* d_ws, size_t ws_size,
                              hipStream_t stream) {
    const int* x      = (const int*)d_in[0];
    float*     out    = (float*)d_out;
    unsigned*  ghist  = (unsigned*)d_ws;

    const int B = out_size / (LEVELS * LEVELS);   // 64
    const int N = in_sizes[0] / B;                // 786432

    // 1) zero the accumulator (B*256 u32)
    hist_zero_kernel<<<dim3((B * LEVELS + 255) / 256), 256, 0, stream>>>(ghist);

    // 2) histogram: 64 blocks per row -> 4096 blocks, ~12K elements each
    const int blocks_per_row = 64;
    int chunk = (N + blocks_per_row - 1) / blocks_per_row;
    chunk = (chunk + 1023) & ~1023;               // multiple of 256 threads * 4 elems
    hist_kernel<<<dim3(blocks_per_row, B), 256, 0, stream>>>(x, ghist, N, chunk);

    // 3) broadcast via WMMA: 16 M-tiles per row = 4 blocks x 4 waves
    bcast_wmma_kernel<<<dim3(4, B), 128, 0, stream>>>(ghist, out);
}